// GSTDP_LIF_NeuronModel_5514738008437
// MI455X (gfx1250) — compile-verified
//
#include <hip/hip_runtime.h>

#define NN 8192
#define TILE_R 32
#define TILE_C 2048
#define TPB 256

typedef float f4 __attribute__((ext_vector_type(4)));

// ---------------------------------------------------------------------------
// CDNA5 async copy: global -> LDS, 16 bytes per active lane, tracked by
// ASYNCcnt.  lds_addr is the 32-bit LDS byte address (low 32 bits of the
// flat shared pointer), gaddr a per-lane 64-bit global address.
// ---------------------------------------------------------------------------
__device__ __forceinline__ void async_load_b128(unsigned lds_addr, const float* gaddr) {
    asm volatile("global_load_async_to_lds_b128 %0, %1, off"
                 :: "v"(lds_addr), "v"(gaddr) : "memory");
}
__device__ __forceinline__ void wait_asynccnt0() {
    asm volatile("s_wait_asynccnt 0" ::: "memory");
}

// ---------------------------------------------------------------------------
// Kernel A: LIF fire step over N neurons (tiny, bandwidth-irrelevant).
// ---------------------------------------------------------------------------
__global__ __launch_bounds__(TPB) void lif_fire_kernel(
    const int*   __restrict__ in_spikes,
    const float* __restrict__ v_in,
    const float* __restrict__ r_in,
    float* __restrict__ out_spk,
    float* __restrict__ out_v,
    float* __restrict__ out_r)
{
    int i = blockIdx.x * TPB + threadIdx.x;
    if (i >= NN) return;
    float v = v_in[i] + (float)in_spikes[i] * 1.0f;   // THRESHOLD = 1.0
    float r = fmaxf(r_in[i] - 1.0f, 0.0f);
    bool  m = (v >= 1.0f) && (r == 0.0f);
    float s = m ? 1.0f : 0.0f;
    v = m ? 0.0f : v;
    r = r + s * 5.0f;                                  // TAU_REFRACT
    out_spk[i] = s;
    out_v[i]   = v;
    out_r[i]   = r;
}

// ---------------------------------------------------------------------------
// Kernel B: streaming STDP weight update.
//   Tile = 32 rows x 2048 cols per block (256 threads, 8 cols/thread).
//   Spike segments staged into LDS via global_load_async_to_lds_b128 and
//   reused across all 32 rows.  Weights streamed with NT 128-bit accesses.
// ---------------------------------------------------------------------------
__global__ __launch_bounds__(TPB) void stdp_update_kernel(
    const float* __restrict__ w_in,
    const float* __restrict__ spk,     // spikes written by kernel A
    float* __restrict__ w_out)
{
    __shared__ float col_spk[TILE_C];
    __shared__ float row_spk[TILE_R];

    const int t        = threadIdx.x;
    const int tilesPerRow = NN / TILE_C;            // 4
    const int tileCol  = blockIdx.x & (tilesPerRow - 1);
    const int tileRow  = blockIdx.x / tilesPerRow;
    const int col0     = tileCol * TILE_C;
    const int row0     = tileRow * TILE_R;

    // --- async stage spike segments into LDS (ASYNCcnt path) ---
    {
        unsigned l0 = (unsigned)(size_t)&col_spk[t * 4];
        async_load_b128(l0, spk + col0 + t * 4);
        unsigned l1 = (unsigned)(size_t)&col_spk[1024 + t * 4];
        async_load_b128(l1, spk + col0 + 1024 + t * 4);
        if (t < TILE_R / 4) {
            unsigned l2 = (unsigned)(size_t)&row_spk[t * 4];
            async_load_b128(l2, spk + row0 + t * 4);
        }
    }
    wait_asynccnt0();
    __syncthreads();

    const int j0 = col0 + t * 8;                    // 8 consecutive cols/thread

    // column spikes + col indices: invariant across the 32 rows
    f4 cs0 = *(const f4*)&col_spk[t * 8];
    f4 cs1 = *(const f4*)&col_spk[t * 8 + 4];
    float cs[8] = {cs0.x, cs0.y, cs0.z, cs0.w, cs1.x, cs1.y, cs1.z, cs1.w};
    float fj[8];
#pragma unroll
    for (int k = 0; k < 8; ++k) fj[k] = (float)(j0 + k);

    for (int rr = 0; rr < TILE_R; ++rr) {
        const int    i    = row0 + rr;
        const float  s_i  = row_spk[rr];            // wave-uniform
        const size_t base = (size_t)i * NN + j0;

        f4 w0 = __builtin_nontemporal_load((const f4*)(w_in + base));
        f4 w1 = __builtin_nontemporal_load((const f4*)(w_in + base + 4));
        float w[8] = {w0.x, w0.y, w0.z, w0.w, w1.x, w1.y, w1.z, w1.w};

        if (s_i > 0.0f) {                           // uniform branch: skip exp for silent rows
            const float fi = (float)i;
#pragma unroll
            for (int k = 0; k < 8; ++k) {
                float d   = fj[k] - fi;             // delta_t = j - i
                float ltp = 0.01f * __expf(d * d * (-1.0f / 800.0f)); // ALPHA_PLUS, 2*TAU_PLUS^2
                w[k] += ((cs[k] > 0.0f) && (d > 0.0f)) ? ltp : 0.0f;
            }
        }
#pragma unroll
        for (int k = 0; k < 8; ++k)
            w[k] = fminf(fmaxf(w[k], 0.0f), 1.0f);  // clip(0,1) on full matrix

        f4 o0 = {w[0], w[1], w[2], w[3]};
        f4 o1 = {w[4], w[5], w[6], w[7]};
        __builtin_nontemporal_store(o0, (f4*)(w_out + base));
        __builtin_nontemporal_store(o1, (f4*)(w_out + base + 4));
    }
}

// ---------------------------------------------------------------------------
// Launch: outputs concatenated flat as [spikes | new_weights | v | r].
// ---------------------------------------------------------------------------
extern "C" void kernel_launch(void* const* d_in, const int* in_sizes, int n_in,
                              void* d_out, int out_size, void* d_ws, size_t ws_size,
                              hipStream_t stream) {
    const int*   in_spikes = (const int*)  d_in[0];
    const float* weights   = (const float*)d_in[1];
    const float* membrane  = (const float*)d_in[2];
    const float* refract   = (const float*)d_in[3];

    float* out     = (float*)d_out;
    float* out_spk = out;
    float* out_w   = out + NN;
    float* out_v   = out + NN + (size_t)NN * NN;
    float* out_r   = out_v + NN;

    lif_fire_kernel<<<NN / TPB, TPB, 0, stream>>>(
        in_spikes, membrane, refract, out_spk, out_v, out_r);

    const int nblocks = (NN / TILE_R) * (NN / TILE_C);   // 1024
    stdp_update_kernel<<<nblocks, TPB, 0, stream>>>(
        weights, out_spk, out_w);
}